// E_STFGNN_21320217658121
// MI455X (gfx1250) — compile-verified
//
#include <hip/hip_runtime.h>
#include <hip/hip_bf16.h>
#include <math.h>

// ---------------------------------------------------------------------------
// E-STFGNN forward for MI455X (gfx1250, wave32).
// Dominant op: M = A_f @ H  ([4096x4096] x [4096x1536], f32) -> f32 WMMA
// (v_wmma_f32_16x16x4_f32). 64x64 register-blocked tile per wave (4x4
// fragments): 16 FLOP/byte from L2, halving L2 traffic vs 32x32 tiles.
// A_f (64MB) + H (24MB) fit in the 192MB L2, so the GEMM streams from L2.
// ---------------------------------------------------------------------------

#define N_NODES 4096
#define T_STEPS 24
#define D_MODEL 64
#define KD_DIM  32
#define H2_DIM  32
#define TOPK_K  8
#define NNZ_CNT 32768
#define NC_COLS (T_STEPS * D_MODEL)   // 1536

typedef __attribute__((ext_vector_type(2))) float v2f;
typedef __attribute__((ext_vector_type(8))) float v8f;

// ---------------------------------------------------------------------------
// K1: edge/weather embed MLPs + combine -> H [N,T,64]. One wave per (n,t) row.
// ---------------------------------------------------------------------------
__global__ __launch_bounds__(256)
void embed_kernel(const float* __restrict__ Xe, const float* __restrict__ Xw,
                  const float* __restrict__ eW1, const float* __restrict__ eb1,
                  const float* __restrict__ eW2, const float* __restrict__ eb2,
                  const float* __restrict__ wW1, const float* __restrict__ wb1,
                  const float* __restrict__ wW2, const float* __restrict__ wb2,
                  const float* __restrict__ Wc,  const float* __restrict__ bc,
                  float* __restrict__ H) {
  __shared__ float lds[8][192];  // per wave: he[32] hw[32] xe[64] xw[64]
  const int w    = threadIdx.x >> 5;
  const int lane = threadIdx.x & 31;
  const long row = (long)blockIdx.x * 8 + w;          // row in [0, N*T)
  const float* xe = Xe + row * 16;
  const float* xw = Xw + row * 8;

  float he = eb1[lane];
#pragma unroll
  for (int i = 0; i < 16; ++i) he += xe[i] * eW1[i * 32 + lane];
  he = fmaxf(he, 0.f);
  float hw = wb1[lane];
#pragma unroll
  for (int i = 0; i < 8; ++i) hw += xw[i] * wW1[i * 32 + lane];
  hw = fmaxf(hw, 0.f);
  lds[w][lane]      = he;
  lds[w][32 + lane] = hw;
  __syncthreads();

#pragma unroll
  for (int rep = 0; rep < 2; ++rep) {
    const int d = lane + rep * 32;
    float ae = eb2[d], aw = wb2[d];
    for (int k = 0; k < 32; ++k) {
      ae += lds[w][k]      * eW2[k * 64 + d];
      aw += lds[w][32 + k] * wW2[k * 64 + d];
    }
    lds[w][64 + d]  = ae;   // edge embedding (no relu: MLP output is linear)
    lds[w][128 + d] = aw;   // weather embedding
  }
  __syncthreads();

#pragma unroll
  for (int rep = 0; rep < 2; ++rep) {
    const int d = lane + rep * 32;
    float a = bc[d];
    for (int k = 0; k < 64; ++k) {
      a += lds[w][64 + k]  * Wc[k * 64 + d];
      a += lds[w][128 + k] * Wc[(64 + k) * 64 + d];
    }
    H[row * 64 + d] = fmaxf(a, 0.f);
  }
}

// ---------------------------------------------------------------------------
// K2: Z = mean_t H  [N,64]
// ---------------------------------------------------------------------------
__global__ void zmean_kernel(const float* __restrict__ H, float* __restrict__ Z) {
  const int i = blockIdx.x * 256 + threadIdx.x;       // over N*64
  if (i >= N_NODES * D_MODEL) return;
  const int n = i >> 6, d = i & 63;
  float s = 0.f;
  for (int t = 0; t < T_STEPS; ++t) s += H[((long)n * T_STEPS + t) * 64 + d];
  Z[i] = s * (1.f / (float)T_STEPS);
}

// ---------------------------------------------------------------------------
// K3: Q = Z@Wq, K = Z@Wk   [N,32]
// ---------------------------------------------------------------------------
__global__ void qk_kernel(const float* __restrict__ Z,
                          const float* __restrict__ Wq, const float* __restrict__ Wk,
                          float* __restrict__ Q, float* __restrict__ Km) {
  const int i = blockIdx.x * 256 + threadIdx.x;       // over N*32
  if (i >= N_NODES * KD_DIM) return;
  const int n = i >> 5, j = i & 31;
  const float* z = Z + n * 64;
  float q = 0.f, k = 0.f;
  for (int d = 0; d < 64; ++d) {
    const float zv = z[d];
    q += zv * Wq[d * 32 + j];
    k += zv * Wk[d * 32 + j];
  }
  Q[i] = q;
  Km[i] = k;
}

// ---------------------------------------------------------------------------
// K4: per-row S = relu(tanh(Q K^T)), exact top-8 threshold (8 argmax rounds,
// tie-exact vs reference "S >= kth"), row-normalize, write (1-gte)*A_t -> A_f.
// One block (256 threads) per row, S row staged in LDS.
// ---------------------------------------------------------------------------
__global__ __launch_bounds__(256)
void topk_kernel(const float* __restrict__ Q, const float* __restrict__ Km,
                 const float* __restrict__ alpha_p, float* __restrict__ Af) {
  __shared__ float sRow[N_NODES];
  __shared__ float redV[256];
  __shared__ int   redI[256];
  __shared__ float winV[TOPK_K];
  __shared__ int   winI[TOPK_K];
  __shared__ float s_kth;
  const int n = blockIdx.x;
  const int tid = threadIdx.x;

  float q[KD_DIM];
#pragma unroll
  for (int i = 0; i < KD_DIM; ++i) q[i] = Q[n * KD_DIM + i];

  for (int m = tid; m < N_NODES; m += 256) {
    float acc = 0.f;
#pragma unroll
    for (int i = 0; i < KD_DIM; ++i) acc += q[i] * Km[m * KD_DIM + i];
    const float t = tanhf(acc);
    sRow[m] = t > 0.f ? t : 0.f;
  }
  __syncthreads();

  for (int r = 0; r < TOPK_K; ++r) {
    float bv = -1.f; int bi = 0;
    for (int m = tid; m < N_NODES; m += 256) {
      const float v = sRow[m];
      if (v > bv) { bv = v; bi = m; }
    }
    redV[tid] = bv; redI[tid] = bi;
    __syncthreads();
    for (int s = 128; s > 0; s >>= 1) {
      if (tid < s && redV[tid + s] > redV[tid]) {
        redV[tid] = redV[tid + s]; redI[tid] = redI[tid + s];
      }
      __syncthreads();
    }
    if (tid == 0) {
      winV[r] = redV[0]; winI[r] = redI[0];
      sRow[redI[0]] = -1.f;                 // remove one instance (tie-safe)
    }
    __syncthreads();
  }
  if (tid == 0) {
    for (int r = 0; r < TOPK_K; ++r) sRow[winI[r]] = winV[r];  // restore
    s_kth = winV[TOPK_K - 1];
  }
  __syncthreads();
  const float kth = s_kth;

  float ps = 0.f;
  for (int m = tid; m < N_NODES; m += 256) {
    float v = sRow[m];
    if (v < kth) v = 0.f;                   // keep S >= kth
    sRow[m] = v;
    ps += v;
  }
  redV[tid] = ps;
  __syncthreads();
  for (int s = 128; s > 0; s >>= 1) {
    if (tid < s) redV[tid] += redV[tid + s];
    __syncthreads();
  }
  const float gte = 1.f / (1.f + __expf(-alpha_p[0]));
  const float scale = (1.f - gte) / (redV[0] + 1e-8f);
  for (int m = tid; m < N_NODES; m += 256)
    Af[(size_t)n * N_NODES + m] = sRow[m] * scale;
}

// ---------------------------------------------------------------------------
// K5: sparse adjacency: degree, then gte * D^-1 A scatter-add into A_f
// ---------------------------------------------------------------------------
__global__ void zero_kernel(float* __restrict__ p, int n) {
  const int i = blockIdx.x * 256 + threadIdx.x;
  if (i < n) p[i] = 0.f;
}
__global__ void degree_kernel(const int* __restrict__ rows,
                              const float* __restrict__ vals,
                              float* __restrict__ deg) {
  const int i = blockIdx.x * 256 + threadIdx.x;
  if (i < NNZ_CNT) atomicAdd(&deg[rows[i]], vals[i]);
}
__global__ void scatter_kernel(const int* __restrict__ rows, const int* __restrict__ cols,
                               const float* __restrict__ vals, const float* __restrict__ deg,
                               const float* __restrict__ alpha_p, float* __restrict__ Af) {
  const int i = blockIdx.x * 256 + threadIdx.x;
  if (i >= NNZ_CNT) return;
  const float gte = 1.f / (1.f + __expf(-alpha_p[0]));
  const int r = rows[i], c = cols[i];
  atomicAdd(&Af[(size_t)r * N_NODES + c], gte * vals[i] / (deg[r] + 1e-8f));
}

// ---------------------------------------------------------------------------
// K6: WMMA f32 GEMM  C[4096,1536] = A[4096,4096] @ B[4096,1536]
// One wave -> 64x64 tile (4x4 fragments of 16x16), K-loop step 4.
// A frag (16x4):  lane l holds A[l&15][2*(l>>4)+{0,1}]  -> aligned float2 load
// B frag (4x16):  lane l holds B[2*(l>>4)+{0,1}][l&15]
// D frag (16x16): vgpr r <-> row r + 8*(l>>4), col l&15
// 16 WMMA (32768 FLOP) per 2KB of fragments -> 16 FLOP/byte from L2.
// ---------------------------------------------------------------------------
__global__ __launch_bounds__(128)
void gemm_wmma_kernel(const float* __restrict__ A, const float* __restrict__ B,
                      float* __restrict__ C) {
  const int NK = N_NODES;
  const int NC = NC_COLS;
  const int wid  = blockIdx.x * 4 + (threadIdx.x >> 5);
  const int tc   = wid % (NC / 64);
  const int tr   = wid / (NC / 64);
  const int r0   = tr * 64, c0 = tc * 64;
  const int lane = threadIdx.x & 31;
  const int half = lane >> 4;
  const int m    = lane & 15;

  v8f acc[4][4] = {{{}}};
  const float* Ap = A + (size_t)(r0 + m) * NK + half * 2;
  const float* Bp = B + c0 + m;

  for (int k0 = 0; k0 < NK; k0 += 4) {
    v2f a[4], b[4];
#pragma unroll
    for (int i = 0; i < 4; ++i)
      a[i] = *(const v2f*)(Ap + (size_t)(16 * i) * NK + k0);
    const float* bp = Bp + (size_t)(k0 + half * 2) * NC;
#pragma unroll
    for (int j = 0; j < 4; ++j) {
      b[j].x = bp[16 * j];
      b[j].y = bp[NC + 16 * j];
    }
#pragma unroll
    for (int i = 0; i < 4; ++i) {
#pragma unroll
      for (int j = 0; j < 4; ++j) {
        acc[i][j] = __builtin_amdgcn_wmma_f32_16x16x4_f32(
            false, a[i], false, b[j], (short)0, acc[i][j], false, false);
      }
    }
  }

  const int rb = r0 + half * 8;
  const int cb = c0 + m;
#pragma unroll
  for (int i = 0; i < 4; ++i) {
#pragma unroll
    for (int j = 0; j < 4; ++j) {
#pragma unroll
      for (int r = 0; r < 8; ++r) {
        C[(size_t)(rb + 16 * i + r) * NC + cb + 16 * j] = acc[i][j][r];
      }
    }
  }
}

// ---------------------------------------------------------------------------
// K7: spatial step  H = relu(M @ Ws + bs) + H   (in place, per (row,d) thread)
// ---------------------------------------------------------------------------
__global__ __launch_bounds__(256)
void spatial_kernel(const float* __restrict__ M, const float* __restrict__ Ws,
                    const float* __restrict__ bs, float* __restrict__ H) {
  const long i = (long)blockIdx.x * 256 + threadIdx.x;   // over N*T*64
  const long row = i >> 6;
  const int d = (int)(i & 63);
  const float* mr = M + row * 64;
  float a = bs[d];
#pragma unroll 8
  for (int k = 0; k < 64; ++k) a += mr[k] * Ws[k * 64 + d];
  H[i] = fmaxf(a, 0.f) + H[i];
}

// ---------------------------------------------------------------------------
// K8: temporal conv (k=3, SAME, channels D->D) + residual, node tile in LDS
// c[t,o] = sum_i sum_kh H[t+kh-1,i] * Wt[o,i,kh];  H = relu(c+bt) + H
// ---------------------------------------------------------------------------
__global__ __launch_bounds__(256)
void temporal_kernel(const float* __restrict__ Wt, const float* __restrict__ bt,
                     float* __restrict__ H) {
  __shared__ float Hl[T_STEPS * D_MODEL];
  const long n = blockIdx.x;
  float* Hn = H + n * (long)(T_STEPS * D_MODEL);
  for (int i = threadIdx.x; i < T_STEPS * D_MODEL; i += 256) Hl[i] = Hn[i];
  __syncthreads();
  for (int idx = threadIdx.x; idx < T_STEPS * D_MODEL; idx += 256) {
    const int t = idx >> 6, o = idx & 63;
    float c = 0.f;
#pragma unroll
    for (int kh = 0; kh < 3; ++kh) {
      const int tt = t + kh - 1;
      if (tt < 0 || tt >= T_STEPS) continue;
      const float* hrow = &Hl[tt * 64];
      const float* wr = Wt + (long)o * 192 + kh;        // Wt[o][i][kh]
#pragma unroll 8
      for (int i2 = 0; i2 < 64; ++i2) c += hrow[i2] * wr[i2 * 3];
    }
    Hn[idx] = fmaxf(c + bt[o], 0.f) + Hl[idx];
  }
}

// ---------------------------------------------------------------------------
// K9: head  out[n] = relu(H[n,T-1,:] @ W1 + b1) @ W2 + b2   (wave per node)
// ---------------------------------------------------------------------------
__global__ __launch_bounds__(256)
void head_kernel(const float* __restrict__ H, const float* __restrict__ W1,
                 const float* __restrict__ b1, const float* __restrict__ W2,
                 const float* __restrict__ b2, float* __restrict__ out) {
  const int n = blockIdx.x * 8 + (threadIdx.x >> 5);
  const int j = threadIdx.x & 31;
  const float* h = H + ((long)n * T_STEPS + (T_STEPS - 1)) * D_MODEL;
  float a = b1[j];
  for (int d = 0; d < 64; ++d) a += h[d] * W1[d * 32 + j];
  a = fmaxf(a, 0.f) * W2[j];
  for (int off = 16; off > 0; off >>= 1) a += __shfl_xor(a, off, 32);
  if (j == 0) out[n] = a + b2[0];
}

// ---------------------------------------------------------------------------
// Launch. Input order assumption: top-level dict in insertion order
// (X_edges, X_weather, A_s_indices, A_s_values), then params as JAX tree
// leaves (dict keys sorted, capitals first):
//   Wk, Wq, alpha, blocks[0]{Ws,Wt,bs,bt}, blocks[1]{Ws,Wt,bs,bt},
//   combine(Wc,bc), edge_embed(W1,b1,W2,b2), head(W1,b1,W2,b2),
//   weather_embed(W1,b1,W2,b2)
// Workspace (floats): H 6.29M | M 6.29M | Af 16.78M | Z | Q | K | deg  ~120MB
// ---------------------------------------------------------------------------
extern "C" void kernel_launch(void* const* d_in, const int* in_sizes, int n_in,
                              void* d_out, int out_size, void* d_ws, size_t ws_size,
                              hipStream_t stream) {
  const float* Xe    = (const float*)d_in[0];
  const float* Xw    = (const float*)d_in[1];
  const int*   Aidx  = (const int*)d_in[2];     // [2, NNZ]: rows then cols
  const float* Avals = (const float*)d_in[3];
  const float* Wk    = (const float*)d_in[4];
  const float* Wq    = (const float*)d_in[5];
  const float* alpha = (const float*)d_in[6];
  const float* Wsb[2] = { (const float*)d_in[7],  (const float*)d_in[11] };
  const float* Wtb[2] = { (const float*)d_in[8],  (const float*)d_in[12] };
  const float* bsb[2] = { (const float*)d_in[9],  (const float*)d_in[13] };
  const float* btb[2] = { (const float*)d_in[10], (const float*)d_in[14] };
  const float* Wc  = (const float*)d_in[15];
  const float* bc  = (const float*)d_in[16];
  const float* eW1 = (const float*)d_in[17];
  const float* eb1 = (const float*)d_in[18];
  const float* eW2 = (const float*)d_in[19];
  const float* eb2 = (const float*)d_in[20];
  const float* hW1 = (const float*)d_in[21];
  const float* hb1 = (const float*)d_in[22];
  const float* hW2 = (const float*)d_in[23];
  const float* hb2 = (const float*)d_in[24];
  const float* wW1 = (const float*)d_in[25];
  const float* wb1 = (const float*)d_in[26];
  const float* wW2 = (const float*)d_in[27];
  const float* wb2 = (const float*)d_in[28];

  float* ws  = (float*)d_ws;
  float* H   = ws;                                        // [N,T,64]
  float* M   = H + (size_t)N_NODES * NC_COLS;             // [N,T,64]
  float* Af  = M + (size_t)N_NODES * NC_COLS;             // [N,N]
  float* Z   = Af + (size_t)N_NODES * N_NODES;            // [N,64]
  float* Q   = Z + (size_t)N_NODES * D_MODEL;             // [N,32]
  float* Km  = Q + (size_t)N_NODES * KD_DIM;              // [N,32]
  float* deg = Km + (size_t)N_NODES * KD_DIM;             // [N]

  // 1) embeddings + combine
  embed_kernel<<<(N_NODES * T_STEPS) / 8, 256, 0, stream>>>(
      Xe, Xw, eW1, eb1, eW2, eb2, wW1, wb1, wW2, wb2, Wc, bc, H);
  // 2) temporal pooling + Q/K
  zmean_kernel<<<(N_NODES * D_MODEL + 255) / 256, 256, 0, stream>>>(H, Z);
  qk_kernel<<<(N_NODES * KD_DIM + 255) / 256, 256, 0, stream>>>(Z, Wq, Wk, Q, Km);
  // 3) learned adjacency (writes full A_f = (1-gte)*A_t)
  topk_kernel<<<N_NODES, 256, 0, stream>>>(Q, Km, alpha, Af);
  // 4) sparse adjacency: A_f += gte * D^-1 A_s
  zero_kernel<<<(N_NODES + 255) / 256, 256, 0, stream>>>(deg, N_NODES);
  degree_kernel<<<(NNZ_CNT + 255) / 256, 256, 0, stream>>>(Aidx, Avals, deg);
  scatter_kernel<<<(NNZ_CNT + 255) / 256, 256, 0, stream>>>(
      Aidx, Aidx + NNZ_CNT, Avals, deg, alpha, Af);
  // 5) fusion blocks
  const int gemm_waves = (N_NODES / 64) * (NC_COLS / 64);  // 1536
  for (int b = 0; b < 2; ++b) {
    gemm_wmma_kernel<<<gemm_waves / 4, 128, 0, stream>>>(Af, H, M);
    spatial_kernel<<<(N_NODES * NC_COLS) / 256, 256, 0, stream>>>(M, Wsb[b], bsb[b], H);
    temporal_kernel<<<N_NODES, 256, 0, stream>>>(Wtb[b], btb[b], H);
  }
  // 6) head
  head_kernel<<<N_NODES / 8, 256, 0, stream>>>(H, hW1, hb1, hW2, hb2, (float*)d_out);
}